// DGM2_O_15985868276158
// MI455X (gfx1250) — compile-verified
//
#include <hip/hip_runtime.h>

typedef float v2f __attribute__((ext_vector_type(2)));
typedef float v8f __attribute__((ext_vector_type(8)));

#define T_STEPS 60
#define D_IN    36
#define L_DIM   10
#define C_NUM   20
#define S_DIM   9
#define OUT_DIM 8
#define KCAT    46   // L + D
#define KEMIT   30   // C + L

__device__ __forceinline__ v8f wmma4(v2f a, v2f b, v8f c) {
  // D = A(16x4,f32) * B(4x16,f32) + C(16x16,f32), exact fp32 matrix op
  return __builtin_amdgcn_wmma_f32_16x16x4_f32(false, a, false, b, (short)0, c,
                                               false, false);
}

// Branch-free fast transcendentals (v_exp_f32 / v_rcp_f32)
__device__ __forceinline__ float fexp(float x)  { return __builtin_amdgcn_exp2f(x * 1.4426950408889634f); }
__device__ __forceinline__ float frcp(float x)  { return __builtin_amdgcn_rcpf(x); }
__device__ __forceinline__ float fsig(float x)  { return frcp(1.f + fexp(-x)); }
__device__ __forceinline__ float ftanh(float x) { return 1.f - 2.f * frcp(1.f + fexp(2.f * x)); }

__global__ __launch_bounds__(32)
void dgm2_scan_kernel(const float* __restrict__ data,
                      const float* __restrict__ ts,
                      const float* __restrict__ sdata,
                      const float* __restrict__ W_update, const float* __restrict__ b_update,
                      const float* __restrict__ W_reset,  const float* __restrict__ b_reset,
                      const float* __restrict__ W_new,    const float* __restrict__ b_new,
                      const float* __restrict__ W_emit,   const float* __restrict__ b_emit,
                      const float* __restrict__ W_ode,    const float* __restrict__ b_ode,
                      const float* __restrict__ W_mlp,    const float* __restrict__ b_mlp,
                      float* __restrict__ out)
{
  // Per-wave tiles (one wave per block => workgroup barriers lower to S_NOP)
  __shared__ float YC[16 * 12];   // cols 0..9: state / yi / yi*r / y_new ; cols 10..11: x0,x1
  __shared__ float ZIN[16 * 32];  // cols 0..19: y_prob ; cols 20..29: y_new ; 30..31: zero
  __shared__ float ZLOG[16 * 20]; // emit logits

  const int lane = threadIdx.x;
  const int n    = lane & 15;   // column (B/C/D layout) or row m (A layout)
  const int hi   = lane >> 4;
  const int koff = hi * 2;      // K offset for lanes 16..31
  const int b0   = blockIdx.x * 16;

  for (int i = lane; i < 16 * 12; i += 32) YC[i]  = 0.f;
  for (int i = lane; i < 16 * 32; i += 32) ZIN[i] = 0.f;
  __syncthreads();

  // ---- pack weights into WMMA B-operand chunks (register resident) ----
  // B layout: VGPR v: lanes 0-15 -> row k0+v ; lanes 16-31 -> row k0+2+v
  // Branch-free: clamped-address load + select.
  auto ldB = [&](const float* W, int K, int N, int k0, int col) -> v2f {
    v2f r;
#pragma unroll
    for (int v = 0; v < 2; ++v) {
      int k  = k0 + koff + v;
      int kc = (k < K) ? k : 0;
      int cc = (col < N) ? col : 0;
      float val = W[kc * N + cc];
      r[v] = (k < K && col < N) ? val : 0.f;
    }
    return r;
  };

  v2f Wu[12], Wr[12], Wn[12], Wo[3], We0[8], We1[8];
#pragma unroll
  for (int c = 0; c < 12; ++c) {
    Wu[c] = ldB(W_update, KCAT, L_DIM, 4 * c, n);
    Wr[c] = ldB(W_reset,  KCAT, L_DIM, 4 * c, n);
    Wn[c] = ldB(W_new,    KCAT, L_DIM, 4 * c, n);
  }
#pragma unroll
  for (int c = 0; c < 3; ++c) Wo[c] = ldB(W_ode, L_DIM, L_DIM, 4 * c, n);
#pragma unroll
  for (int c = 0; c < 8; ++c) {
    We0[c] = ldB(W_emit, KEMIT, C_NUM, 4 * c, n);
    We1[c] = ldB(W_emit, KEMIT, C_NUM, 4 * c, 16 + n);
  }

  const float bU  = (n < L_DIM) ? b_update[n] : 0.f;
  const float bR  = (n < L_DIM) ? b_reset[n]  : 0.f;
  const float bN  = (n < L_DIM) ? b_new[n]    : 0.f;
  const float bO  = (n < L_DIM) ? b_ode[n]    : 0.f;
  const float bE0 = b_emit[n];
  const float bE1 = (n < C_NUM - 16) ? b_emit[16 + n] : 0.f;
  const float bM  = (n < OUT_DIM) ? b_mlp[n] : 0.f;

  const v8f z8 = {0.f, 0.f, 0.f, 0.f, 0.f, 0.f, 0.f, 0.f};
  v8f y_c = z8;        // state in C/D layout
  v8f acc_out = z8;    // running MLP output (C/D layout, cols 0..7 used)
  float t_prev = 0.f;

  for (int t = 0; t < T_STEPS; ++t) {
    const float tcur = ts[t];
    const float dt = (t == 0) ? 0.01f : (tcur - t_prev);
    t_prev = tcur;

    const float* xrow = data + ((size_t)(b0 + n) * T_STEPS + t) * D_IN;

    // stage x0,x1 so the mixed K-chunk (cols 8..11) comes from one LDS b64
    if (lane < 16) *(v2f*)&YC[lane * 12 + 10] = *(const v2f*)xrow;

    // x A-chunks for K = 12..47 (x indices 2..37). Tail chunk is branch-free:
    // lanes hi=1 of chunk 8 would be x[36..37] (padding) -> load safe addr, select 0.
    v2f xc[9];
#pragma unroll
    for (int j = 0; j < 8; ++j) {
      int k = 2 + 4 * j + koff;
      xc[j] = *(const v2f*)(xrow + k);
    }
    {
      v2f tl = *(const v2f*)(xrow + 34);            // valid for all lanes
      xc[8][0] = hi ? 0.f : tl[0];
      xc[8][1] = hi ? 0.f : tl[1];
    }

    // ---- ODE Euler step: yi = y + tanh(y @ W_ode + b) * dt ----
    v2f aY[3];
#pragma unroll
    for (int c = 0; c < 3; ++c) aY[c] = *(const v2f*)&YC[n * 12 + 4 * c + koff];
    v8f g = z8;
#pragma unroll
    for (int c = 0; c < 3; ++c) g = wmma4(aY[c], Wo[c], g);
    v8f yi;
#pragma unroll
    for (int v = 0; v < 8; ++v) yi[v] = y_c[v] + ftanh(g[v] + bO) * dt;

    if (n < L_DIM) {
#pragma unroll
      for (int v = 0; v < 8; ++v) YC[(v + 8 * hi) * 12 + n] = yi[v];
    }
    __syncthreads();

    // ---- update / reset gates (shared A operand, 4 concurrent WMMA chains) ----
#pragma unroll
    for (int c = 0; c < 3; ++c) aY[c] = *(const v2f*)&YC[n * 12 + 4 * c + koff];
    v8f au0 = z8, au1 = z8, ar0 = z8, ar1 = z8;
    au0 = wmma4(aY[0], Wu[0], au0);  ar0 = wmma4(aY[0], Wr[0], ar0);
    au1 = wmma4(aY[1], Wu[1], au1);  ar1 = wmma4(aY[1], Wr[1], ar1);
    au0 = wmma4(aY[2], Wu[2], au0);  ar0 = wmma4(aY[2], Wr[2], ar0);
#pragma unroll
    for (int j = 0; j < 9; ++j) {
      if (j & 1) { au1 = wmma4(xc[j], Wu[3 + j], au1); ar1 = wmma4(xc[j], Wr[3 + j], ar1); }
      else       { au0 = wmma4(xc[j], Wu[3 + j], au0); ar0 = wmma4(xc[j], Wr[3 + j], ar0); }
    }
    v8f us, rs;
#pragma unroll
    for (int v = 0; v < 8; ++v) {
      us[v] = fsig(au0[v] + au1[v] + bU);
      rs[v] = fsig(ar0[v] + ar1[v] + bR);
    }

    if (n < L_DIM) {
#pragma unroll
      for (int v = 0; v < 8; ++v) YC[(v + 8 * hi) * 12 + n] = yi[v] * rs[v];
    }
    __syncthreads();

    // ---- candidate gate (2 concurrent chains) ----
#pragma unroll
    for (int c = 0; c < 3; ++c) aY[c] = *(const v2f*)&YC[n * 12 + 4 * c + koff];
    v8f an0 = z8, an1 = z8;
    an0 = wmma4(aY[0], Wn[0], an0);
    an1 = wmma4(aY[1], Wn[1], an1);
    an0 = wmma4(aY[2], Wn[2], an0);
#pragma unroll
    for (int j = 0; j < 9; ++j) {
      if (j & 1) an1 = wmma4(xc[j], Wn[3 + j], an1);
      else       an0 = wmma4(xc[j], Wn[3 + j], an0);
    }
    v8f ynew;
#pragma unroll
    for (int v = 0; v < 8; ++v) {
      float nv = an0[v] + an1[v] + bN;
      ynew[v] = (1.f - us[v]) * nv + us[v] * yi[v];
    }

    if (n < L_DIM) {
#pragma unroll
      for (int v = 0; v < 8; ++v) {
        YC[(v + 8 * hi) * 12 + n] = ynew[v];
        ZIN[(v + 8 * hi) * 32 + 20 + n] = ynew[v];
      }
    }
    __syncthreads();

    // ---- fold states_t @ W_mlp[10t:10t+10] into the output accumulator ----
    // Branch-free B-chunk build: clamped-address load + select.
#pragma unroll
    for (int c = 0; c < 3; ++c) aY[c] = *(const v2f*)&YC[n * 12 + 4 * c + koff];
#pragma unroll
    for (int c = 0; c < 3; ++c) {
      v2f bm;
#pragma unroll
      for (int v = 0; v < 2; ++v) {
        int k  = 4 * c + koff + v;
        int kc = (k < L_DIM) ? k : 0;
        float val = W_mlp[(t * L_DIM + kc) * OUT_DIM + (n & 7)];
        bm[v] = (k < L_DIM && n < OUT_DIM) ? val : 0.f;
      }
      acc_out = wmma4(aY[c], bm, acc_out);
    }

    // ---- emit: softmax([y_prob | y_new] @ W_emit + b), N=20 as two N-tiles ----
    v8f e0 = z8, e1 = z8;
#pragma unroll
    for (int c = 0; c < 8; ++c) {
      v2f a = *(const v2f*)&ZIN[n * 32 + 4 * c + koff];
      e0 = wmma4(a, We0[c], e0);
      e1 = wmma4(a, We1[c], e1);
    }
#pragma unroll
    for (int v = 0; v < 8; ++v) ZLOG[(v + 8 * hi) * 20 + n] = e0[v] + bE0;
    if (n < 4) {
#pragma unroll
      for (int v = 0; v < 8; ++v) ZLOG[(v + 8 * hi) * 20 + 16 + n] = e1[v] + bE1;
    }
    __syncthreads();

    if (lane < 16) {
      float mx = ZLOG[lane * 20];
#pragma unroll
      for (int c = 1; c < C_NUM; ++c) mx = fmaxf(mx, ZLOG[lane * 20 + c]);
      float e[C_NUM]; float s = 0.f;
#pragma unroll
      for (int c = 0; c < C_NUM; ++c) { e[c] = fexp(ZLOG[lane * 20 + c] - mx); s += e[c]; }
      float inv = frcp(s);
#pragma unroll
      for (int c = 0; c < C_NUM; ++c) ZIN[lane * 32 + c] = e[c] * inv;
    }
    y_c = ynew;
    __syncthreads();
  }

  // ---- static tail: static_data @ W_mlp[600:609] (branch-free guards) ----
#pragma unroll
  for (int c = 0; c < 3; ++c) {
    v2f a, bm;
#pragma unroll
    for (int v = 0; v < 2; ++v) {
      int k  = 4 * c + koff + v;
      int kc = (k < S_DIM) ? k : 0;
      float sv = sdata[(size_t)(b0 + n) * S_DIM + kc];
      float wv = W_mlp[(T_STEPS * L_DIM + kc) * OUT_DIM + (n & 7)];
      a[v]  = (k < S_DIM) ? sv : 0.f;
      bm[v] = (k < S_DIM && n < OUT_DIM) ? wv : 0.f;
    }
    acc_out = wmma4(a, bm, acc_out);
  }

  // store: C/D layout -> out[b0 + v + 8*hi][n], cols < 8
  if (n < OUT_DIM) {
#pragma unroll
    for (int v = 0; v < 8; ++v) {
      int row = b0 + v + 8 * hi;
      out[(size_t)row * OUT_DIM + n] = acc_out[v] + bM;
    }
  }
}

extern "C" void kernel_launch(void* const* d_in, const int* in_sizes, int n_in,
                              void* d_out, int out_size, void* d_ws, size_t ws_size,
                              hipStream_t stream) {
  const float* data     = (const float*)d_in[0];
  const float* ts       = (const float*)d_in[1];
  const float* sdata    = (const float*)d_in[2];
  const float* W_update = (const float*)d_in[3];
  const float* b_update = (const float*)d_in[4];
  const float* W_reset  = (const float*)d_in[5];
  const float* b_reset  = (const float*)d_in[6];
  const float* W_new    = (const float*)d_in[7];
  const float* b_new    = (const float*)d_in[8];
  const float* W_emit   = (const float*)d_in[9];
  const float* b_emit   = (const float*)d_in[10];
  const float* W_ode    = (const float*)d_in[11];
  const float* b_ode    = (const float*)d_in[12];
  const float* W_mlp    = (const float*)d_in[13];
  const float* b_mlp    = (const float*)d_in[14];
  float* out = (float*)d_out;

  const int B = in_sizes[0] / (T_STEPS * D_IN);   // 16384
  dim3 grid(B / 16), block(32);
  hipLaunchKernelGGL(dgm2_scan_kernel, grid, block, 0, stream,
                     data, ts, sdata,
                     W_update, b_update, W_reset, b_reset, W_new, b_new,
                     W_emit, b_emit, W_ode, b_ode, W_mlp, b_mlp, out);
}